// LimitedAdaptiveConv1d_83064667505012
// MI455X (gfx1250) — compile-verified
//
#include <hip/hip_runtime.h>

// ---------------------------------------------------------------------------
// LimitedAdaptiveConv1d for MI455X (gfx1250, wave32)
//   Stage 1: per-frame filter synthesis via V_WMMA_F32_16X16X4_F32
//            (B operands staged in LDS, branch-free inner loop)
//   Stage 2: time-domain 16-tap adaptive FIR with crossfaded coefficients
// ---------------------------------------------------------------------------

typedef float v2f __attribute__((ext_vector_type(2)));
typedef float v8f __attribute__((ext_vector_type(8)));

#define NB      32
#define NF      500
#define FEAT    256
#define KTAP    16
#define FRAME   160
#define OVL     40
#define NS      (NF * FRAME)           // 80000
#define NROWS   (NB * NF)              // 16000 frames total
#define NGRP    (NROWS / 16)           // 1000 wave-groups of 16 frames

#define BSTRIDE 260                    // padded LDS row stride (bank-conflict free)

#define GAIN_A  0.6907755278982137f    // (log_max - log_min)/2, log scale
#define GAIN_B  0.0f                   // (log_max + log_min)/2
#define SHAPE_G 0.5011872336272722f    // 10^(-6/20)
#define ID_TAP  7                      // K - 1 - K/2

// ---------------------------------------------------------------------------
// Kernel 1: filter generation.
//   GEMM: (16000 x 256) @ (256 x [64 ck | 2 gain | 14 pad]) with f32 WMMA,
//   then L2-normalize / identity-blend / gain-scale per (frame, cout).
//   Output wbuf: [row=b*NF+f][o][i][k]  (64 floats per frame).
// ---------------------------------------------------------------------------
__global__ __launch_bounds__(256)
void filter_gen_kernel(const float* __restrict__ feat,   // (16000, 256)
                       const float* __restrict__ ckw,    // (64, 256)
                       const float* __restrict__ ckb,    // (64)
                       const float* __restrict__ fgw,    // (2, 256)
                       const float* __restrict__ fgb,    // (2)
                       float* __restrict__ wbuf)         // (16000, 64)
{
    // 80 rows x 260 floats = 83.2 KB; reused as per-wave 16x80 C-spill later.
    __shared__ float wlds[80 * BSTRIDE];

    const int tid = threadIdx.x;

    // --- Stage B block into LDS: rows 0-63 = ck_w, 64-65 = fg_w, 66-79 = 0.
    for (int e = tid; e < 80 * FEAT; e += 256) {
        int r = e >> 8;                 // e / 256
        int c = e & 255;                // e % 256
        float v = 0.0f;
        if (r < 64)      v = ckw[r * FEAT + c];
        else if (r < 66) v = fgw[(r - 64) * FEAT + c];
        wlds[r * BSTRIDE + c] = v;
    }
    __syncthreads();

    const int lane = tid & 31;
    const int wid  = tid >> 5;                     // wave in block (0..7)
    const int grp  = blockIdx.x * 8 + wid;         // 16-frame group id
    const int hf   = lane >> 4;                    // lane half (K split)
    const int ln   = lane & 15;                    // M or N index

    // A fragment source: 16 feature rows of this group, K offset 2*hf.
    const float* arow = feat + (size_t)(grp * 16 + ln) * FEAT + 2 * hf;
    __builtin_prefetch(arow, 0, 1);                // global_prefetch_b8
    __builtin_prefetch(arow + 128, 0, 1);

    // B fragment sources (LDS): B[k][n] = W[n][k], n striped over lanes.
    const float* b0p = &wlds[(0 * 16 + ln) * BSTRIDE + 2 * hf];
    const float* b1p = &wlds[(1 * 16 + ln) * BSTRIDE + 2 * hf];
    const float* b2p = &wlds[(2 * 16 + ln) * BSTRIDE + 2 * hf];
    const float* b3p = &wlds[(3 * 16 + ln) * BSTRIDE + 2 * hf];
    const float* b4p = &wlds[(4 * 16 + ln) * BSTRIDE + 2 * hf];

    v8f acc0 = {}, acc1 = {}, acc2 = {}, acc3 = {}, acc4 = {};

#pragma unroll 4
    for (int k0 = 0; k0 < FEAT; k0 += 4) {
        v2f a  = *(const v2f*)(arow + k0);
        v2f b0 = *(const v2f*)(b0p + k0);
        v2f b1 = *(const v2f*)(b1p + k0);
        v2f b2 = *(const v2f*)(b2p + k0);
        v2f b3 = *(const v2f*)(b3p + k0);
        v2f b4 = *(const v2f*)(b4p + k0);

        acc0 = __builtin_amdgcn_wmma_f32_16x16x4_f32(false, a, false, b0,
                                                     (short)0, acc0, false, false);
        acc1 = __builtin_amdgcn_wmma_f32_16x16x4_f32(false, a, false, b1,
                                                     (short)0, acc1, false, false);
        acc2 = __builtin_amdgcn_wmma_f32_16x16x4_f32(false, a, false, b2,
                                                     (short)0, acc2, false, false);
        acc3 = __builtin_amdgcn_wmma_f32_16x16x4_f32(false, a, false, b3,
                                                     (short)0, acc3, false, false);
        acc4 = __builtin_amdgcn_wmma_f32_16x16x4_f32(false, a, false, b4,
                                                     (short)0, acc4, false, false);
    }

    // Weights no longer needed: reuse LDS as per-wave 16x80 C-spill tiles.
    __syncthreads();
    float* my = wlds + wid * (16 * 80);
#pragma unroll
    for (int r = 0; r < 8; ++r) {
        my[(r + 8 * hf) * 80 +  0 + ln] = acc0[r];
        my[(r + 8 * hf) * 80 + 16 + ln] = acc1[r];
        my[(r + 8 * hf) * 80 + 32 + ln] = acc2[r];
        my[(r + 8 * hf) * 80 + 48 + ln] = acc3[r];
        my[(r + 8 * hf) * 80 + 64 + ln] = acc4[r];
    }
    __syncthreads();

    // Postprocess: lane -> (local frame mf, cout o).
    const int mf = lane >> 1;
    const int o  = lane & 1;
    const float* rowp = my + mf * 80;

    float glog = rowp[64 + o] + fgb[o];
    float gain = __expf(GAIN_A * tanhf(glog) + GAIN_B);

    float v[32];
    float ss = 0.0f;
#pragma unroll
    for (int j = 0; j < 32; ++j) {
        float t = rowp[o * 32 + j] + ckb[o * 32 + j];
        v[j] = t;
        ss += t * t;
    }
    float inv = 1.0f / (1e-6f + sqrtf(ss));

    const int row = grp * 16 + mf;                  // global frame index
    float* wout = wbuf + ((size_t)row * 2 + o) * 32;
#pragma unroll
    for (int j = 0; j < 32; ++j) {
        float wv = SHAPE_G * v[j] * inv
                 + (((j & 15) == ID_TAP) ? (1.0f - SHAPE_G) : 0.0f);
        wout[j] = gain * wv;
    }
}

// ---------------------------------------------------------------------------
// Kernel 2: adaptive FIR + overlap-add crossfade.
//   One block per (b, f): 320 threads = 2 couts x 160 samples.
//   out[b,o,f*160+t] = sum_{i,k} c[i,k] * x[b,i,f*160+t-k]
//   with c = win1[t]*w_f + win2[t]*w_{f-1} in the first 40 samples.
// ---------------------------------------------------------------------------
__global__ __launch_bounds__(320)
void adaptive_fir_kernel(const float* __restrict__ x,     // (32, 2, 80000)
                         const float* __restrict__ wbuf,  // (16000, 64)
                         const float* __restrict__ owin,  // (40)
                         float* __restrict__ out)         // (32, 2, 80000)
{
    __shared__ float xt[2][176];       // x window: [f*160-15, f*160+160)
    __shared__ float wc[64];           // current-frame coefficients
    __shared__ float wp[64];           // previous-frame coefficients

    const int blk = blockIdx.x;        // b*NF + f
    const int b   = blk / NF;
    const int f   = blk - b * NF;
    const int tid = threadIdx.x;

    // Stage x window (zero-padded at sequence edges).
    for (int j = tid; j < 352; j += 320) {
        int i  = j / 176;
        int jj = j - i * 176;
        int g  = f * FRAME - (KTAP - 1) + jj;
        float val = 0.0f;
        if (g >= 0 && g < NS)
            val = x[((size_t)b * 2 + i) * NS + g];
        xt[i][jj] = val;
    }
    // Stage coefficients.
    if (tid < 64) {
        wc[tid] = wbuf[(size_t)blk * 64 + tid];
        wp[tid] = (f > 0) ? wbuf[(size_t)(blk - 1) * 64 + tid] : 0.0f;
    }
    __syncthreads();

    const int o = (tid >= FRAME) ? 1 : 0;
    const int s = tid - o * FRAME;     // output sample within frame, 0..159

    float cw[32];
    if (s < OVL) {
        float w1 = owin[OVL - 1 - s];            // flipped window (fade-in)
        float w2 = (f > 0) ? owin[s] : 0.0f;     // fade-out of previous frame
#pragma unroll
        for (int j = 0; j < 32; ++j)
            cw[j] = w1 * wc[o * 32 + j] + w2 * wp[o * 32 + j];
    } else {
#pragma unroll
        for (int j = 0; j < 32; ++j)
            cw[j] = wc[o * 32 + j];
    }

    float acc = 0.0f;
#pragma unroll
    for (int i = 0; i < 2; ++i)
#pragma unroll
        for (int k = 0; k < KTAP; ++k)
            acc += cw[i * KTAP + k] * xt[i][s + (KTAP - 1) - k];

    out[((size_t)b * 2 + o) * NS + f * FRAME + s] = acc;
}

// ---------------------------------------------------------------------------
extern "C" void kernel_launch(void* const* d_in, const int* in_sizes, int n_in,
                              void* d_out, int out_size, void* d_ws, size_t ws_size,
                              hipStream_t stream)
{
    const float* x    = (const float*)d_in[0];   // (32, 2, 80000)
    const float* feat = (const float*)d_in[1];   // (32, 500, 256)
    const float* ckw  = (const float*)d_in[2];   // (64, 256)
    const float* ckb  = (const float*)d_in[3];   // (64)
    const float* fgw  = (const float*)d_in[4];   // (2, 256)
    const float* fgb  = (const float*)d_in[5];   // (2)
    const float* owin = (const float*)d_in[6];   // (40)
    float* out  = (float*)d_out;                 // (32, 2, 80000)
    float* wbuf = (float*)d_ws;                  // (16000, 64) scratch filters

    // Stage 1: 1000 wave-groups, 8 waves (256 threads) per block.
    filter_gen_kernel<<<NGRP / 8, 256, 0, stream>>>(feat, ckw, ckb, fgw, fgb, wbuf);

    // Stage 2: one block per (b, f).
    adaptive_fir_kernel<<<NB * NF, 320, 0, stream>>>(x, wbuf, owin, out);
}